// AttentionContextEncoder_3126736192049
// MI455X (gfx1250) — compile-verified
//
#include <hip/hip_runtime.h>
#include <math.h>

// CDNA5 / gfx1250: wave32, WMMA 16x16x4 f32 (exact shape match for K=4 GEMMs)
typedef __attribute__((ext_vector_type(2))) float v2f;
typedef __attribute__((ext_vector_type(8))) float v8f;

// gfx1250 hardware V_TANH_F32 (TRANS op) — confirmed lowering last round.
__device__ __forceinline__ float fast_tanh(float x) {
#if __has_builtin(__builtin_amdgcn_tanhf)
    return __builtin_amdgcn_tanhf(x);
#elif __has_builtin(__builtin_amdgcn_tanh_f32)
    return __builtin_amdgcn_tanh_f32(x);
#else
    float r;
    asm("v_tanh_f32 %0, %1\n\tv_nop" : "=v"(r) : "v"(x));
    return r;
#endif
}

#define NUM_ENT 64
#define DIM_ENT 4
#define HALF    128
#define BATCH   512

// One block per batch element. 512 threads = 16 wave32 waves.
// LDS: p (64x128 f32 = 32KB) + dist (64x64 f32 = 16KB) + xy (512B) ~= 48.5KB.
__global__ __launch_bounds__(512) void ace_fused(
    const float* __restrict__ ctx,   // [NUM_ENT*DIM_ENT, BATCH] = [256, 512]
    const float* __restrict__ Wp,    // [4, 128]
    const float* __restrict__ bp,    // [128]
    const float* __restrict__ Wr,    // [5, 128]
    const float* __restrict__ br,    // [128]
    float* __restrict__ out)         // [BATCH, NUM_ENT, 2*HALF]
{
    __shared__ __align__(16) float p_s[NUM_ENT * HALF];       // p = E_b @ Wr[0:4,:]
    __shared__ __align__(16) float dist_s[NUM_ENT * NUM_ENT]; // pairwise xy distance (symmetric)
    __shared__ float xs[NUM_ENT], ys[NUM_ENT];

    const int b    = blockIdx.x;
    const int tid  = threadIdx.x;
    const int lane = tid & 31;
    // Wave index in an SGPR -> provably uniform tile indices / scalar branches,
    // EXEC stays all-1s around the WMMAs.
    const int wave = __builtin_amdgcn_readfirstlane(tid >> 5);

    // entity x/y coords (ents[b,n,d] = ctx[(n*4+d)*BATCH + b])
    if (tid < NUM_ENT) {
        xs[tid] = ctx[(tid * 4 + 0) * BATCH + b];
        ys[tid] = ctx[(tid * 4 + 1) * BATCH + b];
    }

    // ---- WMMA phase: E_b(64x4) @ [Wp | Wr(0:4,:)] (4x256) ----------------
    // 64 tiles of 16x16x4 (4 M-tiles x 16 N-tiles), 4 tiles per wave.
    // A layout (ISA 7.12.2, 32-bit 16x4): lanes 0-15 -> {K0,K1}, lanes 16-31 -> {K2,K3}.
    const int mrow = lane & 15;
    const int kb   = (lane >> 4) << 1;     // 0 or 2
    const int ncol = lane & 15;
    float* outb = out + (size_t)b * (NUM_ENT * 2 * HALF);

    for (int tt = wave; tt < 64; tt += 16) {          // uniform (SGPR) loop
        const int mt = tt & 3;                        // M-tile (entity rows)
        const int ct = tt >> 2;                       // N-tile (output cols)
        const int n0 = mt * 16;
        const int c0 = ct * 16;
        const int n  = n0 + mrow;

        v2f a, bf;
        a.x = ctx[(n * 4 + kb    ) * BATCH + b];
        a.y = ctx[(n * 4 + kb + 1) * BATCH + b];

        const float* W  = (ct < 8) ? Wp : Wr;         // scalar select
        const int    cc = (ct < 8) ? c0 : (c0 - HALF);
        bf.x = W[(kb    ) * HALF + cc + ncol];
        bf.y = W[(kb + 1) * HALF + cc + ncol];

        v8f c = {};
        c = __builtin_amdgcn_wmma_f32_16x16x4_f32(
                /*neg_a=*/false, a, /*neg_b=*/false, bf,
                /*c_mod=*/(short)0, c, /*reuse_a=*/false, /*reuse_b=*/false);

        // D layout: VGPR v, lanes 0-15 -> M=v, lanes 16-31 -> M=v+8; N = lane&15
        const int mbase = n0 + ((lane >> 4) << 3);
        if (ct < 8) {                                  // scalar branch
            const int   h   = c0 + ncol;
            const float bph = bp[h];
#pragma unroll
            for (int v = 0; v < 8; ++v)
                outb[(mbase + v) * (2 * HALF) + h] = fast_tanh(c[v] + bph);
        } else {
            const int h = cc + ncol;
#pragma unroll
            for (int v = 0; v < 8; ++v)
                p_s[(mbase + v) * HALF + h] = c[v];
        }
    }
    __syncthreads();

    // ---- distance matrix (diagonal = 1.0, matching reference's safe sqrt) ----
    for (int idx = tid; idx < NUM_ENT * NUM_ENT; idx += 512) {
        const int i = idx >> 6, j = idx & 63;
        const float dx = xs[i] - xs[j];
        const float dy = ys[i] - ys[j];
        dist_s[idx] = (i == j) ? 1.0f : sqrtf(dx * dx + dy * dy);
    }
    __syncthreads();

    // ---- pairwise reduction: 4 h per lane (float4), 4 consecutive i per wave.
    // rel_emb[i,h] = sum_{j!=i} tanh((p_i[h]+br[h]) - p_j[h] + dist_ij*Wr4[h])
    // j==i contributes exactly tanh(Wr4[h]+br[h]) (dist_ii=1, p_i-p_i=0): subtract it.
    // dist is symmetric, so dist(i0..i0+3, j) == dist_s[j*64 + i0 .. +3]:
    // one broadcast ds_load_b128 feeds all 4 i-values.
    const int h0 = lane * 4;                          // lane owns h0..h0+3
    const int i0 = wave * 4;                          // wave owns i0..i0+3 (uniform)
    const float4 wr4 = *(const float4*)&Wr[4 * HALF + h0];
    const float4 brv = *(const float4*)&br[h0];
    float4 diag;
    diag.x = fast_tanh(wr4.x + brv.x);
    diag.y = fast_tanh(wr4.y + brv.y);
    diag.z = fast_tanh(wr4.z + brv.z);
    diag.w = fast_tanh(wr4.w + brv.w);

    float4 pb[4], acc[4];
#pragma unroll
    for (int k = 0; k < 4; ++k) {
        const float4 pi = *(const float4*)&p_s[(i0 + k) * HALF + h0];
        pb[k].x = pi.x + brv.x; pb[k].y = pi.y + brv.y;
        pb[k].z = pi.z + brv.z; pb[k].w = pi.w + brv.w;
        acc[k].x = 0.f; acc[k].y = 0.f; acc[k].z = 0.f; acc[k].w = 0.f;
    }

#pragma unroll 2
    for (int j = 0; j < NUM_ENT; ++j) {
        const float4 pj = *(const float4*)&p_s[j * HALF + h0];      // ds_load_b128
        const float4 dj = *(const float4*)&dist_s[j * NUM_ENT + i0]; // b128 broadcast
        const float dk[4] = {dj.x, dj.y, dj.z, dj.w};
#pragma unroll
        for (int k = 0; k < 4; ++k) {
            acc[k].x += fast_tanh(pb[k].x - pj.x + dk[k] * wr4.x);
            acc[k].y += fast_tanh(pb[k].y - pj.y + dk[k] * wr4.y);
            acc[k].z += fast_tanh(pb[k].z - pj.z + dk[k] * wr4.z);
            acc[k].w += fast_tanh(pb[k].w - pj.w + dk[k] * wr4.w);
        }
    }

#pragma unroll
    for (int k = 0; k < 4; ++k) {
        float4 res;
        res.x = acc[k].x - diag.x; res.y = acc[k].y - diag.y;
        res.z = acc[k].z - diag.z; res.w = acc[k].w - diag.w;
        *(float4*)&outb[(i0 + k) * (2 * HALF) + HALF + h0] = res;   // b128 store
    }
}

extern "C" void kernel_launch(void* const* d_in, const int* in_sizes, int n_in,
                              void* d_out, int out_size, void* d_ws, size_t ws_size,
                              hipStream_t stream) {
    const float* ctx = (const float*)d_in[0];  // [256, 512]
    const float* Wp  = (const float*)d_in[1];  // [4, 128]
    const float* bp  = (const float*)d_in[2];  // [128]
    const float* Wr  = (const float*)d_in[3];  // [5, 128]
    const float* br  = (const float*)d_in[4];  // [128]
    float* out = (float*)d_out;                // [512, 64, 256]
    (void)in_sizes; (void)n_in; (void)out_size; (void)d_ws; (void)ws_size;

    ace_fused<<<BATCH, 512, 0, stream>>>(ctx, Wp, bp, Wr, br, out);
}